// GaussianRenderer_51531017617593
// MI455X (gfx1250) — compile-verified
//
#include <hip/hip_runtime.h>
#include <hip/hip_bf16.h>
#include <math.h>

// ---------------------------------------------------------------------------
// Gaussian point renderer for MI455X (gfx1250, wave32).
//   P = 16384 rays x N = 2048 gaussians.
//   rsr/rsm are [P,3]x[3,N] GEMMs -> V_WMMA_F32_16X16X4_F32 (K=3 padded to 4).
//   Transcendental-bound (33.5M rcp+exp); WMMA offloads the dot products.
// ---------------------------------------------------------------------------

#define HH     128
#define WW     128
#define NPTS   2048
#define KSEL   20
#define THRA   0.01f
#define FOCAL  150.0f
#define PPIX   (HH * WW)

typedef float v2f __attribute__((ext_vector_type(2)));
typedef float v8f __attribute__((ext_vector_type(8)));

// ---------------------------------------------------------------------------
// Kernel 1: per-point precompute.
//  ws[0..3N)  : 2*sigma_{x,y,z}            (B operand of rsr GEMM)
//  ws[3N..6N) : 2*sigma_{x,y,z}*w_{x,y,z}  (B operand of rsm GEMM)
//  ws[6N..7N) : mm = m^T S m (x2 folded)
// ---------------------------------------------------------------------------
__global__ void gr_prep_points(const float* __restrict__ verts,
                               const float* __restrict__ sigmas,
                               const float* __restrict__ R,
                               const float* __restrict__ T,
                               float* __restrict__ ws) {
  int n = blockIdx.x * blockDim.x + threadIdx.x;
  if (n >= NPTS) return;
  float v0 = verts[n * 3 + 0], v1 = verts[n * 3 + 1], v2 = verts[n * 3 + 2];
  float s0 = sigmas[n * 3 + 0], s1 = sigmas[n * 3 + 1], s2 = sigmas[n * 3 + 2];
  // m = v @ R + T   (row-vector convention)
  float m0 = v0 * R[0] + v1 * R[3] + v2 * R[6] + T[0];
  float m1 = v0 * R[1] + v1 * R[4] + v2 * R[7] + T[1];
  float m2 = v0 * R[2] + v1 * R[5] + v2 * R[8] + T[2];
  // w = m @ R^T : w_k = sum_j m_j * R[k*3+j]
  float w0 = m0 * R[0] + m1 * R[1] + m2 * R[2];
  float w1 = m0 * R[3] + m1 * R[4] + m2 * R[5];
  float w2 = m0 * R[6] + m1 * R[7] + m2 * R[8];
  float bs0 = 2.f * s0, bs1 = 2.f * s1, bs2 = 2.f * s2;
  float bw0 = bs0 * w0, bw1 = bs1 * w1, bw2 = bs2 * w2;
  ws[0 * NPTS + n] = bs0;
  ws[1 * NPTS + n] = bs1;
  ws[2 * NPTS + n] = bs2;
  ws[3 * NPTS + n] = bw0;
  ws[4 * NPTS + n] = bw1;
  ws[5 * NPTS + n] = bw2;
  ws[6 * NPTS + n] = bw0 * w0 + bw1 * w1 + bw2 * w2;  // mm
}

// ---------------------------------------------------------------------------
// Kernel 2: one wave32 block per 32 pixels.
// ---------------------------------------------------------------------------
__launch_bounds__(32)
__global__ void gr_render(const float* __restrict__ ws,
                          const float* __restrict__ R,
                          float* __restrict__ out) {
  const int lane    = threadIdx.x;        // 0..31
  const int col     = lane & 15;          // WMMA column / A row within tile
  const bool hi     = lane >= 16;         // K-pair (2,3) half of the wave
  const int pixBase = blockIdx.x * 32;

  __shared__ float lact[32][17];          // stride 17 -> conflict-free row reads
  __shared__ float llen[32][17];

  // ---- A operands (loop-invariant): per-pixel ray terms -------------------
  // 16x4 f32 A layout: lane L -> row M=L%16; VGPR0=K0(lo)/K2(hi), VGPR1=K1/K3.
  v2f aR[2], aS[2];                       // aR: (u*u) rows, aS: u rows
#pragma unroll
  for (int t = 0; t < 2; ++t) {
    int q = pixBase + t * 16 + col;       // pixel this lane's A-row represents
    float fy = (float)(q >> 7);           // y = q / W   (W = 128)
    float fx = (float)(q & 127);          // x = q % W
    float r0 = (fx + 0.5f - 0.5f * (float)WW) * (1.0f / FOCAL);
    float r1 = (fy + 0.5f - 0.5f * (float)HH) * (1.0f / FOCAL);
    float r2 = 1.0f;
    // u = ray @ R^T
    float u0 = r0 * R[0] + r1 * R[1] + r2 * R[2];
    float u1 = r0 * R[3] + r1 * R[4] + r2 * R[5];
    float u2 = r0 * R[6] + r1 * R[7] + r2 * R[8];
    aS[t].x = hi ? u2 : u0;
    aS[t].y = hi ? 0.f : u1;
    aR[t].x = aS[t].x * aS[t].x;          // (u2^2,0) / (u0^2,u1^2)
    aR[t].y = aS[t].y * aS[t].y;
  }

  // ---- per-lane (= per-pixel) top-K list, sorted descending by act --------
  float actv[KSEL], lenv[KSEL];
  int   idxv[KSEL];
#pragma unroll
  for (int i = 0; i < KSEL; ++i) { actv[i] = 0.f; lenv[i] = 0.f; idxv[i] = -1; }

  const float* bs  = ws;                  // 2*sigma
  const float* bw  = ws + 3 * NPTS;       // 2*sigma*w
  const float* mmv = ws + 6 * NPTS;       // mm

  for (int c = 0; c < NPTS / 16; ++c) {
    int n = c * 16 + col;                 // this lane's point column
    // B operands: lane L -> col N=L%16; VGPR0=K0(lo)/K2(hi), VGPR1=K1/K3=0.
    v2f bR, bS;
    bR.x = bs[(hi ? 2 * NPTS : 0) + n];
    bR.y = hi ? 0.f : bs[NPTS + n];
    bS.x = bw[(hi ? 2 * NPTS : 0) + n];
    bS.y = hi ? 0.f : bw[NPTS + n];
    float mmn = mmv[n];

    v8f zero = {};
    // D = A x B (+0): 16 pixels x 16 points per op; two pixel sub-tiles.
    v8f rsr0 = __builtin_amdgcn_wmma_f32_16x16x4_f32(false, aR[0], false, bR,
                                                     (short)0, zero, false, false);
    v8f rsr1 = __builtin_amdgcn_wmma_f32_16x16x4_f32(false, aR[1], false, bR,
                                                     (short)0, zero, false, false);
    v8f rsm0 = __builtin_amdgcn_wmma_f32_16x16x4_f32(false, aS[0], false, bS,
                                                     (short)0, zero, false, false);
    v8f rsm1 = __builtin_amdgcn_wmma_f32_16x16x4_f32(false, aS[1], false, bS,
                                                     (short)0, zero, false, false);

    // C layout: VGPR r -> row (r + 8*hi), column = lane%16.
    const int rowoff = hi ? 8 : 0;
#pragma unroll
    for (int r = 0; r < 8; ++r) {
      {
        float rr = rsr0[r], sm = rsm0[r];
        float l  = sm * __builtin_amdgcn_rcpf(rr);
        float a  = __expf(-0.5f * (mmn - sm * l));
        lact[rowoff + r][col] = a;
        llen[rowoff + r][col] = l;
      }
      {
        float rr = rsr1[r], sm = rsm1[r];
        float l  = sm * __builtin_amdgcn_rcpf(rr);
        float a  = __expf(-0.5f * (mmn - sm * l));
        lact[16 + rowoff + r][col] = a;
        llen[16 + rowoff + r][col] = l;
      }
    }
    __syncthreads();                      // single-wave block: cheap dscnt wait

    // ---- scan this lane's own pixel row: 16 candidates --------------------
#pragma unroll
    for (int j = 0; j < 16; ++j) {
      float a  = lact[lane][j];
      float ll = llen[lane][j];
      bool want = (a > THRA) && (ll > 0.f) && (a > actv[KSEL - 1]);
      if (want) {                         // register bubble-insert (sorted desc)
        float ca = a, cl = ll;
        int   ci = c * 16 + j;
#pragma unroll
        for (int i = 0; i < KSEL; ++i) {
          bool gt = ca > actv[i];
          float ta = actv[i], tl = lenv[i];
          int   ti = idxv[i];
          actv[i] = gt ? ca : ta;  lenv[i] = gt ? cl : tl;  idxv[i] = gt ? ci : ti;
          ca      = gt ? ta : ca;  cl      = gt ? tl : cl;  ci      = gt ? ti : ci;
        }
      }
    }
    __syncthreads();                      // WAR: before next chunk's LDS writes
  }

  // ---- stable sort selected entries by hit length (invalid -> +inf) -------
  float key[KSEL];
#pragma unroll
  for (int i = 0; i < KSEL; ++i)
    key[i] = (actv[i] > 0.f) ? lenv[i] : __builtin_inff();
#pragma unroll
  for (int pass = 0; pass < KSEL - 1; ++pass) {
#pragma unroll
    for (int j = 0; j < KSEL - 1 - pass; ++j) {
      bool sw = key[j] > key[j + 1];      // strict > : stable (matches argsort)
      float tk = key[j], ta = actv[j], tl = lenv[j];
      int   ti = idxv[j];
      key[j]      = sw ? key[j + 1]  : tk;
      actv[j]     = sw ? actv[j + 1] : ta;
      lenv[j]     = sw ? lenv[j + 1] : tl;
      idxv[j]     = sw ? idxv[j + 1] : ti;
      key[j + 1]  = sw ? tk : key[j + 1];
      actv[j + 1] = sw ? ta : actv[j + 1];
      lenv[j + 1] = sw ? tl : lenv[j + 1];
      idxv[j + 1] = sw ? ti : idxv[j + 1];
    }
  }

  // ---- alpha compositing + outputs ----------------------------------------
  // d_out (flat float): [weight P*K][index P*K][valid P][hit P*K]
  const int p = pixBase + lane;
  float* out_w = out;
  float* out_i = out + PPIX * KSEL;
  float* out_v = out + 2 * PPIX * KSEL;
  float* out_h = out + 2 * PPIX * KSEL + PPIX;

  float excl = 0.f;
  int   cnt  = 0;
#pragma unroll
  for (int i = 0; i < KSEL; ++i) {
    bool  v   = actv[i] > 0.f;
    float wgt = actv[i] * __expf(-excl);  // ABSORB = 1
    excl += actv[i];
    out_w[p * KSEL + i] = wgt;
    out_i[p * KSEL + i] = v ? (float)idxv[i] : -1.0f;
    out_h[p * KSEL + i] = v ? lenv[i] : 0.f;
    cnt += v ? 1 : 0;
  }
  out_v[p] = (float)cnt;
}

// ---------------------------------------------------------------------------
extern "C" void kernel_launch(void* const* d_in, const int* in_sizes, int n_in,
                              void* d_out, int out_size, void* d_ws, size_t ws_size,
                              hipStream_t stream) {
  const float* verts  = (const float*)d_in[0];  // [N,3]
  const float* sigmas = (const float*)d_in[1];  // [N,3]
  const float* R      = (const float*)d_in[2];  // [3,3]
  const float* T      = (const float*)d_in[3];  // [3]
  float* ws  = (float*)d_ws;                    // needs 7*N*4 = 57 KB
  float* out = (float*)d_out;

  gr_prep_points<<<(NPTS + 255) / 256, 256, 0, stream>>>(verts, sigmas, R, T, ws);
  gr_render<<<PPIX / 32, 32, 0, stream>>>(ws, R, out);
}